// DirectionalGCNConv_7602092114065
// MI455X (gfx1250) — compile-verified
//
#include <hip/hip_runtime.h>
#include <math.h>

typedef __attribute__((ext_vector_type(2))) float v2f;
typedef __attribute__((ext_vector_type(8))) float v8f;

// Problem constants: N=64, C=64, T=300, V=25, K=3, E=16, O=64
// x: (64,64,300,25) fp32 ; out: (64,64,300,25) fp32
//
// Workspace layout (floats). Adjacency tiles stored TRANSPOSED + ZERO-PADDED
// as (32 cols x 28 k) so WMMA B-fragments are guard-free aligned b64 loads:
//   [0      .. 57344)  A_effT  (C=64 x [col=32][k=28]), zero padded
//   [57344  .. 57408)  gate    (C=64)
//   [57408  .. 159808) x_mean  (N=64 x C=64 x V=25)
//   [159808 .. 217152) A_dynT  (N=64 x [col=32][k=28]), zero padded

// ---------------------------------------------------------------------------
// Kernel 1: prep — softmax over K of alpha, degree-normalized |A_learned|,
// fold into per-channel A_effT[c][w][v] (transposed, zero-padded); gate[c].
// ---------------------------------------------------------------------------
__global__ __launch_bounds__(256) void prep_kernel(
    const float* __restrict__ A_phys,    // (3,25,25)
    const float* __restrict__ alpha,     // (3,64)
    const float* __restrict__ A_learned, // (25,25)
    const float* __restrict__ alpha_dyn, // (64)
    float* __restrict__ A_effT,          // (64,32,28) padded
    float* __restrict__ gate)            // (64)
{
  __shared__ float aw[3 * 64];
  __shared__ float dinv[25];
  const int tid = threadIdx.x;

  if (tid < 64) {
    float a0 = alpha[tid], a1 = alpha[64 + tid], a2 = alpha[128 + tid];
    float m = fmaxf(a0, fmaxf(a1, a2));
    float e0 = expf(a0 - m), e1 = expf(a1 - m), e2 = expf(a2 - m);
    float inv = 1.0f / (e0 + e1 + e2);
    aw[tid] = e0 * inv; aw[64 + tid] = e1 * inv; aw[128 + tid] = e2 * inv;
    gate[tid] = 1.0f / (1.0f + expf(-alpha_dyn[tid]));
  }
  if (tid < 25) {
    float s = 0.0f;
    for (int w = 0; w < 25; ++w) s += fabsf(A_learned[tid * 25 + w]);
    dinv[tid] = 1.0f / sqrtf(fmaxf(s, 1e-6f));
  }
  __syncthreads();

  for (int idx = tid; idx < 64 * 896; idx += 256) {
    const int c = idx / 896, r = idx % 896;
    const int col = r / 28, kv = r % 28;   // col = w, kv = v
    float val = 0.0f;
    if (col < 25 && kv < 25) {
      const int r25 = kv * 25 + col;       // A[v][w]
      val = dinv[kv] * fabsf(A_learned[r25]) * dinv[col];
      val += aw[c]       * A_phys[r25];
      val += aw[64 + c]  * A_phys[625 + r25];
      val += aw[128 + c] * A_phys[1250 + r25];
    }
    A_effT[idx] = val;
  }
}

// ---------------------------------------------------------------------------
// Kernel 2: x_mean[n,c,v] over T. One block per (n,c); coalesced over v,
// per-thread partial sums then one LDS float atomic (ds_add_f32) each.
// ---------------------------------------------------------------------------
__global__ __launch_bounds__(256) void mean_kernel(
    const float* __restrict__ x, float* __restrict__ xmean)
{
  __shared__ float s[25];
  const int nc = blockIdx.x;
  const int tid = threadIdx.x;
  if (tid < 25) s[tid] = 0.0f;
  __syncthreads();
  const float* xb = x + (size_t)nc * 7500;
  if (tid < 250) {
    const int v = tid % 25;
    float acc = 0.0f;
    for (int j = tid / 25; j < 300; j += 10) acc += xb[j * 25 + v];
    atomicAdd(&s[v], acc);
  }
  __syncthreads();
  if (tid < 25) xmean[nc * 25 + tid] = s[tid] * (1.0f / 300.0f);
}

// ---------------------------------------------------------------------------
// Kernel 3: per-sample dynamic adjacency -> A_dynT[n][w][v] (transposed,
// zero-padded 32x28). One block per n.
// ---------------------------------------------------------------------------
__global__ __launch_bounds__(256) void adyn_kernel(
    const float* __restrict__ xmean,  // (64,64,25)
    const float* __restrict__ Wp,     // (16,64)
    float* __restrict__ AdynT)        // (64,32,28) padded
{
  __shared__ float emb[25 * 16];
  __shared__ float inv[25];
  __shared__ float S[25 * 25];
  const int n = blockIdx.x;
  const int tid = threadIdx.x;

  for (int idx = tid; idx < 896; idx += 256) AdynT[n * 896 + idx] = 0.0f;

  for (int idx = tid; idx < 400; idx += 256) {
    const int v = idx >> 4, e = idx & 15;
    float s = 0.0f;
    for (int c = 0; c < 64; ++c) s += Wp[e * 64 + c] * xmean[(n * 64 + c) * 25 + v];
    emb[v * 16 + e] = s;
  }
  __syncthreads();
  if (tid < 25) {
    float ss = 0.0f;
    for (int e = 0; e < 16; ++e) { float t = emb[tid * 16 + e]; ss += t * t; }
    inv[tid] = 1.0f / fmaxf(sqrtf(ss), 1e-12f);
  }
  __syncthreads();
  for (int idx = tid; idx < 625; idx += 256) {
    const int v = idx / 25, w = idx % 25;
    float s = 0.0f;
    for (int e = 0; e < 16; ++e) s += emb[v * 16 + e] * emb[w * 16 + e];
    S[idx] = s * inv[v] * inv[w];
  }
  __syncthreads();
  if (tid < 25) {
    const int v = tid;
    float m = -1e30f;
    for (int w = 0; w < 25; ++w) m = fmaxf(m, S[v * 25 + w]);
    float sum = 0.0f;
    for (int w = 0; w < 25; ++w) sum += expf(S[v * 25 + w] - m);
    const float r = 1.0f / sum;
    for (int w = 0; w < 25; ++w)
      AdynT[n * 896 + w * 28 + v] = expf(S[v * 25 + w] - m) * r;  // [col=w][k=v]
  }
}

// ---------------------------------------------------------------------------
// Kernel 4: fused main. One block per (t-tile of 16, n). 8 waves.
// Stage 1 (WMMA f32 16x16x4): y = x_tile @ (A_eff[c] + gate[c]*A_dyn[n])
//   A-frags from LDS-staged x (coalesced b128 global loads, zero-padded),
//   B-frags = guard-free aligned b64 loads from transposed padded tiles.
//   y written transposed into LDS ysh[col][c] (stride 66 -> bank stride 2).
// Stage 2 (WMMA f32 16x16x4): out = W_conv @ y, B-frags = single ds b64 each.
// ---------------------------------------------------------------------------
__global__ __launch_bounds__(256) void gcn_main(
    const float* __restrict__ x,      // (64,64,300,25)
    const float* __restrict__ AeffT,  // (64,32,28) padded
    const float* __restrict__ gate,   // (64)
    const float* __restrict__ AdynT,  // (64,32,28) padded
    const float* __restrict__ Wc,     // (64,64)
    float* __restrict__ out)          // (64,64,300,25)
{
  __shared__ float ysh[400 * 66];   // y transposed: [col=tt*25+w][c], 105.6 KB
  __shared__ float xsh[8 * 416];    // per-wave x staging, zero-tailed, 13.3 KB

  const int t0   = blockIdx.x * 16;
  const int n    = blockIdx.y;
  const int tid  = threadIdx.x;
  const int wave = tid >> 5;   // 8 waves (wave32)
  const int lane = tid & 31;
  const int half = lane >> 4;  // K-pair selector in wmma fragments
  const int l16  = lane & 15;  // M/N row-col selector

  float* xs = xsh + wave * 416;
  if (lane < 16) xs[400 + lane] = 0.0f;          // K-tail pad, written once

  const float* AdT = AdynT + n * 896;
  const int rem = 7500 - t0 * 25;                // 400, or 300 on last tile

  // Hoist A_dyn B-fragment loads: identical for every channel c.
  float2 bdyn[2][7];
#pragma unroll
  for (int nt = 0; nt < 2; ++nt) {
    const int col = nt * 16 + l16;
#pragma unroll
    for (int ks = 0; ks < 7; ++ks) {
      const int kb = ks * 4 + half * 2;
      bdyn[nt][ks] = *(const float2*)(AdT + col * 28 + kb);   // aligned b64
    }
  }

  // ---------------- stage 1: per-channel V-contraction -------------------
  for (int c = wave; c < 64; c += 8) {
    const float* slab = x + (size_t)((n * 64 + c) * 300 + t0) * 25; // 400 floats
    if (c + 8 < 64)   // prefetch next channel's slab (global_prefetch_b8)
      __builtin_prefetch((const char*)(slab + 8 * 7500) + lane * 64, 0, 3);

    // Coalesced b128 staging into LDS (zero rows past T boundary).
#pragma unroll
    for (int i0 = 0; i0 < 4; ++i0) {
      const int i = lane * 4 + i0 * 128;
      if (i < 400) {
        float a0 = 0.f, a1 = 0.f, a2 = 0.f, a3 = 0.f;
        if (i < rem) {
          const float4 vv = *(const float4*)(slab + i);   // 16B aligned
          a0 = vv.x; a1 = vv.y; a2 = vv.z; a3 = vv.w;
        }
        xs[i] = a0; xs[i + 1] = a1; xs[i + 2] = a2; xs[i + 3] = a3;
      }
    }
    // Wave-local LDS RAW ordering (lanes read other lanes' stores).
    asm volatile("s_wait_dscnt 0x0" ::: "memory");

    // A fragments from LDS: rows = t (lane&15), K = v padded 25 -> 28.
    v2f a[7];
#pragma unroll
    for (int ks = 0; ks < 7; ++ks) {
      const int kb = ks * 4 + half * 2;
      const float v0 = xs[l16 * 25 + kb];
      const float v1 = xs[l16 * 25 + kb + 1];
      a[ks].x = (kb     < 25) ? v0 : 0.0f;   // folds away except ks==6
      a[ks].y = (kb + 1 < 25) ? v1 : 0.0f;
    }

    const float* AeT = AeffT + c * 896;
    const float gc = gate[c];
#pragma unroll
    for (int nt = 0; nt < 2; ++nt) {
      const int col = nt * 16 + l16;
      v8f acc = {0.f, 0.f, 0.f, 0.f, 0.f, 0.f, 0.f, 0.f};
#pragma unroll
      for (int ks = 0; ks < 7; ++ks) {
        const int kb = ks * 4 + half * 2;
        const float2 be = *(const float2*)(AeT + col * 28 + kb);  // aligned b64
        const float2 bd = bdyn[nt][ks];
        v2f b;
        b.x = fmaf(gc, bd.x, be.x);
        b.y = fmaf(gc, bd.y, be.y);
        acc = __builtin_amdgcn_wmma_f32_16x16x4_f32(
            false, a[ks], false, b, (short)0, acc, false, false);
      }
      if (col < 25) {
#pragma unroll
        for (int i = 0; i < 8; ++i) {
          const int m = i + half * 8;                 // local t row 0..15
          ysh[(m * 25 + col) * 66 + c] = acc[i];      // transposed, pad 66
        }
      }
    }
  }
  __syncthreads();

  // ---------------- stage 2: channel mix with W_conv ----------------------
  const int otile = wave & 3;    // 4 o-tiles of 16
  const int grp   = wave >> 2;   // 2 waves split the 25 col-tiles

  // W_conv A-fragments for this o-tile, aligned b64 loads, held in VGPRs.
  v2f wa[16];
  {
    const int o = otile * 16 + l16;
#pragma unroll
    for (int ks = 0; ks < 16; ++ks) {
      const int kb = ks * 4 + half * 2;
      const float2 ww = *(const float2*)(Wc + o * 64 + kb);
      wa[ks].x = ww.x; wa[ks].y = ww.y;
    }
  }

  const int ntBeg = grp ? 13 : 0;
  const int ntEnd = grp ? 25 : 13;
  for (int nt = ntBeg; nt < ntEnd; ++nt) {
    const int col = nt * 16 + l16;              // 0..399 compacted (t,w)
    const float* yc = ysh + col * 66;
    v8f acc = {0.f, 0.f, 0.f, 0.f, 0.f, 0.f, 0.f, 0.f};
#pragma unroll
    for (int ks = 0; ks < 16; ++ks) {           // K = C = 64, exact
      const int kb = ks * 4 + half * 2;
      const float2 yv = *(const float2*)(yc + kb);   // single ds b64
      v2f b; b.x = yv.x; b.y = yv.y;
      acc = __builtin_amdgcn_wmma_f32_16x16x4_f32(
          false, wa[ks], false, b, (short)0, acc, false, false);
    }
    const int tt = col / 25;
    const int w  = col % 25;
    const int t  = t0 + tt;
    if (t < 300) {
#pragma unroll
      for (int i = 0; i < 8; ++i) {
        const int o = otile * 16 + i + half * 8;
        out[(size_t)((n * 64 + o) * 300 + t) * 25 + w] = acc[i];
      }
    }
  }
}

// ---------------------------------------------------------------------------
extern "C" void kernel_launch(void* const* d_in, const int* in_sizes, int n_in,
                              void* d_out, int out_size, void* d_ws, size_t ws_size,
                              hipStream_t stream) {
  const float* x      = (const float*)d_in[0];
  const float* A_phys = (const float*)d_in[1];
  const float* alpha  = (const float*)d_in[2];
  const float* A_l    = (const float*)d_in[3];
  const float* W_proj = (const float*)d_in[4];
  const float* a_dyn  = (const float*)d_in[5];
  const float* W_conv = (const float*)d_in[6];
  float* out = (float*)d_out;

  float* ws     = (float*)d_ws;
  float* AeffT  = ws + 0;        // 57344
  float* gate   = ws + 57344;    // 64
  float* xmean  = ws + 57408;    // 102400
  float* AdynT  = ws + 159808;   // 57344

  hipLaunchKernelGGL(prep_kernel, dim3(1),       dim3(256), 0, stream,
                     A_phys, alpha, A_l, a_dyn, AeffT, gate);
  hipLaunchKernelGGL(mean_kernel, dim3(64 * 64), dim3(256), 0, stream, x, xmean);
  hipLaunchKernelGGL(adyn_kernel, dim3(64),      dim3(256), 0, stream,
                     xmean, W_proj, AdynT);
  hipLaunchKernelGGL(gcn_main,    dim3(19, 64),  dim3(256), 0, stream,
                     x, AeffT, gate, AdynT, W_conv, out);
}